// GIN_23605140259119
// MI455X (gfx1250) — compile-verified
//
#include <hip/hip_runtime.h>

#define N_NODES 50000
#define N_EDGES 800000
#define FT 128
#define MT 64          // rows per block in MLP kernel
#define LSTR 132       // LDS row stride (floats) - avoids bank conflicts

typedef __attribute__((ext_vector_type(2))) float v2f;
typedef __attribute__((ext_vector_type(8))) float v8f;

// ---------------------------------------------------------------------------
// h[i] = x[i]  (seed the aggregation buffer with x, GIN eps=0)
// ---------------------------------------------------------------------------
__global__ __launch_bounds__(256) void gin_copy(const float* __restrict__ x,
                                                float* __restrict__ h) {
    size_t i = (size_t)blockIdx.x * blockDim.x + threadIdx.x;
    const size_t n4 = (size_t)N_NODES * FT / 4;
    if (i < n4) {
        ((float4*)h)[i] = ((const float4*)x)[i];
    }
}

// ---------------------------------------------------------------------------
// h[dst[e]] += x[src[e]]  : 32 lanes per edge, 4 floats per lane
// ---------------------------------------------------------------------------
__global__ __launch_bounds__(256) void gin_scatter(const float* __restrict__ x,
                                                   const int* __restrict__ src,
                                                   const int* __restrict__ dst,
                                                   float* __restrict__ h) {
    size_t t = (size_t)blockIdx.x * blockDim.x + threadIdx.x;
    size_t e = t >> 5;
    int lane = (int)(t & 31);
    if (e < (size_t)N_EDGES) {
        int s = src[e];
        int d = dst[e];
        float4 v = ((const float4*)(x + (size_t)s * FT))[lane];
        float* p = h + (size_t)d * FT + lane * 4;
        atomicAdd(p + 0, v.x);
        atomicAdd(p + 1, v.y);
        atomicAdd(p + 2, v.z);
        atomicAdd(p + 3, v.w);
    }
}

// ---------------------------------------------------------------------------
// y = relu( relu(h @ W1 + b1) @ W2 + b2 )   for a 64-row tile.
// 8 waves; wave w owns output columns [16w, 16w+16). fp32 WMMA 16x16x4.
// 4 accumulator sets per wave share each B-fragment (4 WMMAs per B load).
// ---------------------------------------------------------------------------
__global__ __launch_bounds__(256) void gin_mlp(const float* __restrict__ h,
                                               const float* __restrict__ W1,
                                               const float* __restrict__ b1,
                                               const float* __restrict__ W2,
                                               const float* __restrict__ b2,
                                               float* __restrict__ y) {
    __shared__ float h1[MT * LSTR];

    const int tid   = threadIdx.x;
    const int wave  = tid >> 5;
    const int lane  = tid & 31;
    const int lhalf = lane >> 4;       // 0 or 1 (which half-wave)
    const int l16   = lane & 15;
    const int n     = wave * 16 + l16; // output column of this lane
    const int rowBlock = blockIdx.x * MT;

    // ---------------- GEMM1: h1 = relu(h_tile @ W1 + b1) -> LDS ------------
    {
        v8f acc[4] = {};
        const float* aptr[4];
#pragma unroll
        for (int m = 0; m < 4; ++m) {
            int r = rowBlock + m * 16 + l16;
            if (r > N_NODES - 1) r = N_NODES - 1;   // clamp: tile either fully
            aptr[m] = h + (size_t)r * FT + 2 * lhalf; // valid or store-masked
        }
        const float* bptr = W1 + (size_t)(2 * lhalf) * FT + n;
#pragma unroll
        for (int k0 = 0; k0 < FT; k0 += 4) {
            v2f b;
            b.x = bptr[(size_t)k0 * FT];
            b.y = bptr[(size_t)k0 * FT + FT];
#pragma unroll
            for (int m = 0; m < 4; ++m) {
                v2f a = *(const v2f*)(aptr[m] + k0);
                acc[m] = __builtin_amdgcn_wmma_f32_16x16x4_f32(
                    false, a, false, b, (short)0, acc[m], false, false);
            }
        }
        float bias = b1[n];
#pragma unroll
        for (int m = 0; m < 4; ++m) {
#pragma unroll
            for (int r = 0; r < 8; ++r) {
                float v = acc[m][r] + bias;
                v = v > 0.0f ? v : 0.0f;
                h1[(m * 16 + r + 8 * lhalf) * LSTR + n] = v;
            }
        }
    }
    __syncthreads();

    // ---------------- GEMM2: y = relu(h1 @ W2 + b2) -> global --------------
    {
        v8f acc[4] = {};
        const float* bptr = W2 + (size_t)(2 * lhalf) * FT + n;
#pragma unroll
        for (int k0 = 0; k0 < FT; k0 += 4) {
            v2f b;
            b.x = bptr[(size_t)k0 * FT];
            b.y = bptr[(size_t)k0 * FT + FT];
#pragma unroll
            for (int m = 0; m < 4; ++m) {
                v2f a = *(const v2f*)(&h1[(m * 16 + l16) * LSTR + 2 * lhalf + k0]);
                acc[m] = __builtin_amdgcn_wmma_f32_16x16x4_f32(
                    false, a, false, b, (short)0, acc[m], false, false);
            }
        }
        float bias = b2[n];
#pragma unroll
        for (int m = 0; m < 4; ++m) {
            int rbase = rowBlock + m * 16;
            if (rbase < N_NODES) {            // uniform guard (N % 16 == 0)
#pragma unroll
                for (int r = 0; r < 8; ++r) {
                    float v = acc[m][r] + bias;
                    v = v > 0.0f ? v : 0.0f;
                    y[(size_t)(rbase + r + 8 * lhalf) * FT + n] = v;
                }
            }
        }
    }
}

// ---------------------------------------------------------------------------
static inline void run_layer(const float* x, const int* src, const int* dst,
                             const float* W1, const float* b1,
                             const float* W2, const float* b2,
                             float* h, float* y, hipStream_t stream) {
    const int copyBlocks = (N_NODES * FT / 4 + 255) / 256;       // 6250
    const int scatBlocks = (int)(((size_t)N_EDGES * 32) / 256);  // 100000
    const int mlpBlocks  = (N_NODES + MT - 1) / MT;              // 782

    gin_copy<<<copyBlocks, 256, 0, stream>>>(x, h);
    gin_scatter<<<scatBlocks, 256, 0, stream>>>(x, src, dst, h);
    gin_mlp<<<mlpBlocks, 256, 0, stream>>>(h, W1, b1, W2, b2, y);
}

extern "C" void kernel_launch(void* const* d_in, const int* in_sizes, int n_in,
                              void* d_out, int out_size, void* d_ws, size_t ws_size,
                              hipStream_t stream) {
    const float* feat = (const float*)d_in[0];
    const int*   ei   = (const int*)d_in[1];
    const float* W1a  = (const float*)d_in[2];
    const float* b1a  = (const float*)d_in[3];
    const float* W2a  = (const float*)d_in[4];
    const float* b2a  = (const float*)d_in[5];
    const float* Ws1  = (const float*)d_in[6];
    const float* bs1  = (const float*)d_in[7];
    const float* Ws2  = (const float*)d_in[8];
    const float* bs2  = (const float*)d_in[9];

    const int* src = ei;
    const int* dst = ei + N_EDGES;

    float* out  = (float*)d_out;
    float* h    = (float*)d_ws;                       // [N_NODES*FT]
    float* xbuf = h + (size_t)N_NODES * FT;           // [N_NODES*FT]

    // Layer 0: feat -> xbuf
    run_layer(feat, src, dst, W1a, b1a, W2a, b2a, h, xbuf, stream);
    // Layer 1: xbuf -> out
    run_layer(xbuf, src, dst, Ws1, bs1, Ws2, bs2, h, out, stream);
    // Layer 2: out -> out  (mlp reads only h; no aliasing hazard)
    run_layer(out, src, dst, Ws1 + FT * FT, bs1 + FT,
              Ws2 + FT * FT, bs2 + FT, h, out, stream);
}